// BEVHDMapFusionNet_81432579932286
// MI455X (gfx1250) — compile-verified
//
#include <hip/hip_runtime.h>
#include <hip/hip_bf16.h>

// ---------------------------------------------------------------------------
// BEV + HD-map fusion net on gfx1250 (MI455X), full f32 via V_WMMA_F32_16X16X4_F32
// Shapes: B=2,T=4 -> BT=8 ; H=W=32 (1024 px) ; F=128 ; Cin1=144 ; Cmap=64
//         Ckv=192 ; heads=4 x d=32 ; Nq=Nk=1024
// ---------------------------------------------------------------------------

typedef __attribute__((ext_vector_type(2))) float v2f;
typedef __attribute__((ext_vector_type(8))) float v8f;

__device__ __forceinline__ v8f wmma4(v2f a, v2f b, v8f c) {
    // D(16x16,f32) = A(16x4,f32) x B(4x16,f32) + C
    return __builtin_amdgcn_wmma_f32_16x16x4_f32(
        /*neg_a=*/false, a, /*neg_b=*/false, b,
        /*c_mod=*/(short)0, c, /*reuse_a=*/false, /*reuse_b=*/false);
}

#define BT   8
#define HW   1024

// ---------------------------------------------------------------------------
// 3x3 SAME conv as 9 shifted channel-GEMMs.  One wave = 16 out-channels x 16 px.
// X: (BT, CIN, 32, 32)   Wt: (128, CIN, 3, 3)   Y: (BT, OCtot, 32, 32) at oc_off
// 4096 waves (8 img * 8 octile * 64 pxtile).  Zero padding via clamped address
// + 0/1 mask multiply (branch-free, no exec-mask save/restore).
// ---------------------------------------------------------------------------
template <int CIN>
__global__ void conv3x3_relu_wmma(const float* __restrict__ X,
                                  const float* __restrict__ Wt,
                                  const float* __restrict__ Bias,
                                  float* __restrict__ Y,
                                  int oc_off, int OCtot) {
    const int wid  = blockIdx.x * (blockDim.x >> 5) + (threadIdx.x >> 5);
    const int lane = threadIdx.x & 31;
    const int n    = lane & 15;         // A: row m  /  B: col n
    const int half = lane >> 4;
    const int koff = half << 1;         // K pair offset per 16x4 f32 layout

    const int img = wid >> 9;           // / (8*64)
    const int ot  = (wid >> 6) & 7;     // output-channel tile (16 ch)
    const int pt  = wid & 63;           // pixel tile (16 px, half a row)
    const int y   = pt >> 1;
    const int xb  = (pt & 1) << 4;

    const float* Xi = X + (size_t)img * CIN * HW;
    const float* Wr = Wt + (size_t)(ot * 16 + n) * CIN * 9;  // this lane's A row

    v8f acc = {};
    for (int r = 0; r < 3; ++r) {
        const int  yy  = y + r - 1;
        const bool yok = (yy >= 0) && (yy < 32);
        const int  yyc = min(max(yy, 0), 31);
        for (int s = 0; s < 3; ++s) {
            const int   xx  = xb + n + s - 1;
            const float msk = (yok && xx >= 0 && xx < 32) ? 1.0f : 0.0f;
            const int   xxc = min(max(xx, 0), 31);
            const int   rs  = r * 3 + s;
            const float* Xp = Xi + yyc * 32 + xxc;   // always in-bounds
#pragma unroll 4
            for (int cb = 0; cb < CIN; cb += 4) {
                const int c0 = cb + koff;
                v2f a, b;
                a.x = Wr[(size_t)c0 * 9 + rs];
                a.y = Wr[(size_t)(c0 + 1) * 9 + rs];
                b.x = Xp[(size_t)c0 * HW] * msk;
                b.y = Xp[(size_t)(c0 + 1) * HW] * msk;
                acc = wmma4(a, b, acc);
            }
        }
    }
#pragma unroll
    for (int v = 0; v < 8; ++v) {
        const int oc = ot * 16 + v + (half << 3);    // D: M = v + 8*half
        float val = acc[v] + Bias[oc];
        val = fmaxf(val, 0.0f);
        Y[((size_t)img * OCtot + oc_off + oc) * HW + y * 32 + xb + n] = val;
    }
}

// ---------------------------------------------------------------------------
// Pointwise channel-major GEMM:  Y[o][n] = sum_c Wt[o][c] * X[c][n]  per image.
// X: (BT, CIN, 1024)   Wt: (128, CIN)   Y: (BT, OCtot, 1024) at oc_off
// 4096 waves.  A pair is contiguous & 8B aligned -> single b64 load.
// ---------------------------------------------------------------------------
template <int CIN>
__global__ void gemm_cm_wmma(const float* __restrict__ X,
                             const float* __restrict__ Wt,
                             const float* __restrict__ Bias,
                             float* __restrict__ Y,
                             int oc_off, int OCtot, int relu) {
    const int wid  = blockIdx.x * (blockDim.x >> 5) + (threadIdx.x >> 5);
    const int lane = threadIdx.x & 31;
    const int n    = lane & 15;
    const int half = lane >> 4;
    const int koff = half << 1;

    const int img = wid >> 9;
    const int ot  = (wid >> 6) & 7;
    const int nb  = (wid & 63) << 4;

    const float* Xi = X + (size_t)img * CIN * HW + nb + n;
    const float* Wr = Wt + (size_t)(ot * 16 + n) * CIN;

    v8f acc = {};
#pragma unroll 8
    for (int cb = 0; cb < CIN; cb += 4) {
        const int c0 = cb + koff;                    // even -> 8B aligned
        v2f a = *(const v2f*)(Wr + c0);              // global_load_b64
        v2f b;
        b.x = Xi[(size_t)c0 * HW];
        b.y = Xi[(size_t)(c0 + 1) * HW];
        acc = wmma4(a, b, acc);
    }
#pragma unroll
    for (int v = 0; v < 8; ++v) {
        const int oc = ot * 16 + v + (half << 3);
        float val = acc[v] + (Bias ? Bias[oc] : 0.0f);
        if (relu) val = fmaxf(val, 0.0f);
        Y[((size_t)img * OCtot + oc_off + oc) * HW + nb + n] = val;
    }
}

// ---------------------------------------------------------------------------
// Flash attention, transposed:  S^T = K * Q^T ;  O^T = V^T * P^T.
// Q/K/V stored channel-major: (img, h*32+d, n).  One wave = 16 queries.
// 2048 waves (8 img * 4 heads * 64 qtiles), 64 key-tiles each.
// Softmax reductions: 8 in-lane ops + one shfl_xor(16) per step.
// ---------------------------------------------------------------------------
__global__ void attention_wmma(const float* __restrict__ Qb,
                               const float* __restrict__ Kb,
                               const float* __restrict__ Vb,
                               float* __restrict__ attnT) {
    const int wid  = blockIdx.x * (blockDim.x >> 5) + (threadIdx.x >> 5);
    const int lane = threadIdx.x & 31;
    const int n    = lane & 15;
    const int half = lane >> 4;

    const int img = wid >> 8;           // / (4*64)
    const int h   = (wid >> 6) & 3;
    const int qb  = (wid & 63) << 4;

    const size_t base = ((size_t)img * 128 + h * 32) * HW;
    const float* Qp = Qb + base;
    const float* Kp = Kb + base;
    const float* Vp = Vb + base;

    // Preload B-operand of the score GEMM: Q^T(d x q) for this query tile.
    v2f qreg[8];
#pragma unroll
    for (int ks = 0; ks < 8; ++ks) {
        const int d0 = ks * 4 + (half << 1);
        qreg[ks].x = Qp[(size_t)d0 * HW + qb + n];
        qreg[ks].y = Qp[(size_t)(d0 + 1) * HW + qb + n];
    }

    const float scale = 0.17677669529663689f;   // 1/sqrt(32)
    float mrun = -1.0e30f, lrun = 0.0f;
    v8f acc0 = {}, acc1 = {};                   // O^T tiles: d 0..15 / 16..31

    for (int kt = 0; kt < 64; ++kt) {
        const int kb = kt << 4;

        // S^T(16k x 16q): A = K tile rows (m=key, K-dim=d)
        v8f s = {};
#pragma unroll
        for (int ks = 0; ks < 8; ++ks) {
            const int d0 = ks * 4 + (half << 1);
            v2f a;
            a.x = Kp[(size_t)d0 * HW + kb + n];
            a.y = Kp[(size_t)(d0 + 1) * HW + kb + n];
            s = wmma4(a, qreg[ks], s);
        }

        // Online softmax per query column (lane column + cross-half swap).
        float tm = -1.0e30f;
#pragma unroll
        for (int v = 0; v < 8; ++v) { s[v] *= scale; tm = fmaxf(tm, s[v]); }
        tm = fmaxf(tm, __shfl_xor(tm, 16, 32));
        const float mnew = fmaxf(mrun, tm);
        const float corr = __expf(mrun - mnew);

        float P[8], psum = 0.0f;
#pragma unroll
        for (int v = 0; v < 8; ++v) { P[v] = __expf(s[v] - mnew); psum += P[v]; }
        psum += __shfl_xor(psum, 16, 32);
        lrun = lrun * corr + psum;
        mrun = mnew;
#pragma unroll
        for (int v = 0; v < 8; ++v) { acc0[v] *= corr; acc1[v] *= corr; }

        float Psw[8];
#pragma unroll
        for (int v = 0; v < 8; ++v) Psw[v] = __shfl_xor(P[v], 16, 32);

        // O^T += V^T * P^T.  B operand needs key k0=4j+2*half (+1) for column q:
        // lane owns keys v+8*half; other half's values come via Psw (same q col).
#pragma unroll
        for (int j = 0; j < 4; ++j) {
            v2f b;
            if (j == 0)      { b.x = half ? Psw[2] : P[0];  b.y = half ? Psw[3] : P[1]; }
            else if (j == 1) { b.x = half ? Psw[6] : P[4];  b.y = half ? Psw[7] : P[5]; }
            else if (j == 2) { b.x = half ? P[2] : Psw[0];  b.y = half ? P[3] : Psw[1]; }
            else             { b.x = half ? P[6] : Psw[4];  b.y = half ? P[7] : Psw[5]; }
            const int k0 = kb + j * 4 + (half << 1);        // even -> 8B aligned
            v2f a0 = *(const v2f*)(Vp + (size_t)n * HW + k0);         // b64 load
            v2f a1 = *(const v2f*)(Vp + (size_t)(16 + n) * HW + k0);  // b64 load
            acc0 = wmma4(a0, b, acc0);
            acc1 = wmma4(a1, b, acc1);
        }
    }

    const float inv_l = 1.0f / lrun;
#pragma unroll
    for (int v = 0; v < 8; ++v) {
        const int d = v + (half << 3);
        attnT[base + (size_t)d * HW + qb + n]        = acc0[v] * inv_l;
        attnT[base + (size_t)(16 + d) * HW + qb + n] = acc1[v] * inv_l;
    }
}

// ---------------------------------------------------------------------------
// Elementwise prep kernels
// ---------------------------------------------------------------------------
__global__ void prep_x1_kernel(const float* __restrict__ bev,
                               const float* __restrict__ ego,
                               float* __restrict__ X1) {
    const int idx = blockIdx.x * blockDim.x + threadIdx.x;
    if (idx >= BT * 144 * HW) return;
    const int img = idx / (144 * HW);
    const int rem = idx - img * 144 * HW;
    const int c   = rem / HW;
    const int p   = rem - c * HW;
    X1[idx] = (c < 128) ? bev[((size_t)img * 128 + c) * HW + p]
                        : ego[img * 16 + (c - 128)];
}

__global__ void prep_ego_x3_kernel(const float* __restrict__ ego,
                                   float* __restrict__ X3) {
    const int idx = blockIdx.x * blockDim.x + threadIdx.x;
    if (idx >= BT * 16 * HW) return;
    const int img = idx / (16 * HW);
    const int rem = idx - img * 16 * HW;
    const int c   = rem / HW;
    const int p   = rem - c * HW;
    X3[((size_t)img * 144 + 128 + c) * HW + p] = ego[img * 16 + c];
}

// Bilinear 16x16 -> 32x32 (half-pixel centers, edge clamp), write into kv ch 128..191
__global__ void front_resize_kernel(const float* __restrict__ front,
                                    float* __restrict__ kv) {
    const int idx = blockIdx.x * blockDim.x + threadIdx.x;
    if (idx >= BT * 64 * HW) return;
    const int img = idx / (64 * HW);
    const int rem = idx - img * 64 * HW;
    const int c   = rem / HW;
    const int p   = rem - c * HW;
    const int y   = p >> 5, x = p & 31;

    const float sy = y * 0.5f - 0.25f;
    const float sx = x * 0.5f - 0.25f;
    const int fy = (int)floorf(sy), fx = (int)floorf(sx);
    const float wy = sy - fy, wx = sx - fx;
    const int y0 = min(max(fy, 0), 15),     x0 = min(max(fx, 0), 15);
    const int y1 = min(max(fy + 1, 0), 15), x1 = min(max(fx + 1, 0), 15);

    const float* F = front + ((size_t)img * 64 + c) * 256;
    const float v00 = F[y0 * 16 + x0], v01 = F[y0 * 16 + x1];
    const float v10 = F[y1 * 16 + x0], v11 = F[y1 * 16 + x1];
    const float val = (1.0f - wy) * ((1.0f - wx) * v00 + wx * v01)
                    +         wy  * ((1.0f - wx) * v10 + wx * v11);
    kv[((size_t)img * 192 + 128 + c) * HW + p] = val;
}

// ---------------------------------------------------------------------------
extern "C" void kernel_launch(void* const* d_in, const int* in_sizes, int n_in,
                              void* d_out, int out_size, void* d_ws, size_t ws_size,
                              hipStream_t stream) {
    const float* bev   = (const float*)d_in[0];
    const float* hdmap = (const float*)d_in[1];
    const float* ego   = (const float*)d_in[2];
    const float* front = (const float*)d_in[3];
    const float* w_bev = (const float*)d_in[4];
    const float* b_bev = (const float*)d_in[5];
    const float* w_hd  = (const float*)d_in[6];
    const float* b_hd  = (const float*)d_in[7];
    const float* wq    = (const float*)d_in[8];
    const float* wk    = (const float*)d_in[9];
    const float* wv    = (const float*)d_in[10];
    const float* wo    = (const float*)d_in[11];
    const float* bo    = (const float*)d_in[12];
    const float* w_out = (const float*)d_in[13];
    const float* b_out = (const float*)d_in[14];
    float* out = (float*)d_out;

    // Workspace layout (floats); X3 aliases X1, attnT aliases bev_feat.
    float* ws = (float*)d_ws;
    float* X1    = ws;                         // 8*144*1024  (also X3)
    float* BF    = X1 + (size_t)BT * 144 * HW; // 8*128*1024  (bev_feat, also attnT)
    float* KV    = BF + (size_t)BT * 128 * HW; // 8*192*1024
    float* Qb    = KV + (size_t)BT * 192 * HW; // 8*128*1024
    float* Kb    = Qb + (size_t)BT * 128 * HW;
    float* Vb    = Kb + (size_t)BT * 128 * HW;
    float* X3    = X1;
    float* attnT = BF;

    const dim3 blk(128);         // 4 waves / block
    const dim3 gemm_grid(1024);  // 4096 waves
    const dim3 attn_grid(512);   // 2048 waves

    // 1. bev ++ ego -> X1 (8,144,32,32)
    prep_x1_kernel<<<(BT * 144 * HW + 255) / 256, 256, 0, stream>>>(bev, ego, X1);
    // 2. conv1: X1 -> bev_feat (relu)
    conv3x3_relu_wmma<144><<<gemm_grid, blk, 0, stream>>>(X1, w_bev, b_bev, BF, 0, 128);
    // 3. conv2: hd_map -> KV channels 0..127 (relu)
    conv3x3_relu_wmma<64><<<gemm_grid, blk, 0, stream>>>(hdmap, w_hd, b_hd, KV, 0, 192);
    // 4. bilinear front -> KV channels 128..191
    front_resize_kernel<<<(BT * 64 * HW + 255) / 256, 256, 0, stream>>>(front, KV);
    // 5-7. projections (channel-major outputs)
    gemm_cm_wmma<128><<<gemm_grid, blk, 0, stream>>>(BF, wq, nullptr, Qb, 0, 128, 0);
    gemm_cm_wmma<192><<<gemm_grid, blk, 0, stream>>>(KV, wk, nullptr, Kb, 0, 128, 0);
    gemm_cm_wmma<192><<<gemm_grid, blk, 0, stream>>>(KV, wv, nullptr, Vb, 0, 128, 0);
    // 8. attention -> attnT (channel-major)
    attention_wmma<<<attn_grid, blk, 0, stream>>>(Qb, Kb, Vb, attnT);
    // 9. output projection (+bo) -> X3 channels 0..127
    gemm_cm_wmma<128><<<gemm_grid, blk, 0, stream>>>(attnT, wo, bo, X3, 0, 144, 0);
    // 10. ego -> X3 channels 128..143
    prep_ego_x3_kernel<<<(BT * 16 * HW + 255) / 256, 256, 0, stream>>>(ego, X3);
    // 11. conv3: X3 -> out (relu)
    conv3x3_relu_wmma<144><<<gemm_grid, blk, 0, stream>>>(X3, w_out, b_out, out, 0, 128);
}